// PatchSampler2d_51651276702082
// MI455X (gfx1250) — compile-verified
//
#include <hip/hip_runtime.h>

// PatchSampler2d for MI455X (gfx1250):
//   per (p,s,c): Y[32x32] = My[32x128] * X[128x128] * Mx^T[128x32]
// - async global->LDS DMA (ASYNCcnt) of the cropped row band
// - pool matrices built in LDS while DMA is in flight
// - both GEMMs on V_WMMA_F32_16X16X4_F32 (f32-exact, wave32)
// - contraction ranges pruned to the crop window (uniform per block)
// - K-loops unrolled 2x so ds-loads of step k+4 overlap WMMAs of step k

constexpr int kP = 32, kS = 4, kC = 64, kH = 128, kW = 128;
constexpr int kNC = 32, kO = 32;
constexpr int kPad = 130;  // padded LDS row stride (floats) -> conflict-free A/B frags

typedef __attribute__((ext_vector_type(2))) float v2f;
typedef __attribute__((ext_vector_type(8))) float v8f;

__device__ __forceinline__ unsigned lds_u32(const void* p) {
  // LDS aperture: low 32 bits of a generic pointer to __shared__ are the LDS byte address
  return (unsigned)(unsigned long long)p;
}

__global__ __launch_bounds__(128) void patch_sampler2d_wmma(
    const float* __restrict__ input, const int* __restrict__ size_raw,
    const int* __restrict__ pix, const int* __restrict__ channels,
    float* __restrict__ out) {
  __shared__ float sX[kH * kW];      // 64 KB raw slice, row stride 128
  __shared__ float sMy[kO * kPad];   // row-pool matrix  (padded)
  __shared__ float sMx[kO * kPad];   // col-pool matrix  (padded)
  __shared__ float sT[kO * kPad];    // intermediate My*X (padded)

  const int b   = blockIdx.x;
  const int p   = b / (kS * kNC);
  const int rem = b - p * (kS * kNC);
  const int s   = rem / kNC;
  const int cp  = rem - s * kNC;

  const int tid   = threadIdx.x;
  const int wave  = tid >> 5;
  const int lane  = tid & 31;
  const int l15   = lane & 15;
  const int khalf = (lane < 16) ? 0 : 2;  // K sub-offset per WMMA A/B layout

  // ---- scalar crop parameters (uniform per block) ---------------------
  const int size   = size_raw[p] + kO;             // MIN_SIZE == 32
  const int wrange = kW - size + 1;
  const int hrange = kH - size + 1;
  const int pm     = pix[p] % (wrange * hrange);
  const int offy   = pm / wrange;
  const int offx   = pm - offy * wrange;
  const int src_c  = channels[p * kNC + cp];

  const int ky0 = offy & ~3;                       // GEMM1 K range (rows of X)
  const int ky1 = (offy + size + 3) & ~3;
  const int kx0 = offx & ~3;                       // GEMM2 K range (cols of X)
  const int kx1 = (offx + size + 3) & ~3;

  // ---- async DMA: needed row band of X, global -> LDS -----------------
  {
    const unsigned long long gbase = (unsigned long long)(
        input + (((unsigned long long)p * kS + s) * kC + src_c) * (kH * kW));
    const unsigned sx_base  = lds_u32(&sX[0]);
    const int nvec4         = (ky1 - ky0) * (kW / 4);  // float4 chunks in band
    const unsigned band0    = (unsigned)ky0 * kW * 4u; // byte offset of band start
    for (int f = tid; f < nvec4; f += 128) {
      const unsigned byteoff = band0 + (unsigned)f * 16u;
      const unsigned ldsaddr = sx_base + byteoff;
      asm volatile("global_load_async_to_lds_b128 %0, %1, %2"
                   :: "v"(ldsaddr), "v"(byteoff), "s"(gbase)
                   : "memory");
    }
  }

  // ---- build adaptive-avg-pool matrices while the DMA flies -----------
  for (int e = tid; e < kO * kH; e += 128) {
    const int o = e >> 7;
    const int r = e & (kH - 1);
    const int sy = offy + ((o * size) >> 5);
    const int ey = offy + (((o + 1) * size + 31) >> 5);
    sMy[o * kPad + r] = (r >= sy && r < ey) ? 1.0f / (float)(ey - sy) : 0.0f;
    const int sx = offx + ((o * size) >> 5);
    const int ex = offx + (((o + 1) * size + 31) >> 5);
    sMx[o * kPad + r] = (r >= sx && r < ex) ? 1.0f / (float)(ex - sx) : 0.0f;
  }

  asm volatile("s_wait_asynccnt 0" ::: "memory");
  __syncthreads();

  // ---- GEMM1: T[32x128] = My * X  (K = rows, pruned to crop band) -----
  // 16 C-tiles (2 Mtiles x 8 Ntiles); each wave: one Mtile x 4 Ntiles.
  const int mtile = wave & 1;
  const int nbase = (wave >> 1) * 64;
  {
    // crop cols [offx, offx+size) never touch this 64-wide chunk? skip it.
    const bool active_chunk = (offx < nbase + 64) && (offx + size > nbase);
    if (active_chunk) {
      v8f a0 = {}, a1 = {}, a2 = {}, a3 = {};
      const float* Ap = &sMy[(mtile * 16 + l15) * kPad + khalf];
      const float* Bp = &sX[khalf * kW + nbase + l15];
#pragma unroll 2
      for (int k = ky0; k < ky1; k += 4) {
        const v2f a = *(const v2f*)(Ap + k);          // one ds_load_b64
        const float* xp = Bp + k * kW;
        v2f b0 = {xp[0],  xp[kW]};
        v2f b1 = {xp[16], xp[16 + kW]};
        v2f b2 = {xp[32], xp[32 + kW]};
        v2f b3 = {xp[48], xp[48 + kW]};
        a0 = __builtin_amdgcn_wmma_f32_16x16x4_f32(false, a, false, b0, (short)0, a0, false, false);
        a1 = __builtin_amdgcn_wmma_f32_16x16x4_f32(false, a, false, b1, (short)0, a1, false, false);
        a2 = __builtin_amdgcn_wmma_f32_16x16x4_f32(false, a, false, b2, (short)0, a2, false, false);
        a3 = __builtin_amdgcn_wmma_f32_16x16x4_f32(false, a, false, b3, (short)0, a3, false, false);
      }
      const int mrow = mtile * 16 + ((lane < 16) ? 0 : 8);
#pragma unroll
      for (int r = 0; r < 8; ++r) {
        float* tp = &sT[(mrow + r) * kPad + nbase + l15];
        tp[0]  = a0[r];
        tp[16] = a1[r];
        tp[32] = a2[r];
        tp[48] = a3[r];
      }
    }
  }
  __syncthreads();

  // ---- GEMM2: Y[32x32] = T * Mx^T  (K = cols, pruned to crop band) ----
  // 4 C-tiles, one per wave. B[k][n] = Mx[n][k]: contiguous b64 frag loads.
  {
    const int ntile = wave >> 1;
    v8f acc = {};
    const float* Ap = &sT[(mtile * 16 + l15) * kPad + khalf];
    const float* Bp = &sMx[(ntile * 16 + l15) * kPad + khalf];
#pragma unroll 2
    for (int k = kx0; k < kx1; k += 4) {
      const v2f a  = *(const v2f*)(Ap + k);
      const v2f bb = *(const v2f*)(Bp + k);
      acc = __builtin_amdgcn_wmma_f32_16x16x4_f32(false, a, false, bb, (short)0, acc, false, false);
    }
    float* op = out + (((unsigned long long)p * kS + s) * kNC + cp) * (kO * kO);
    const int m0 = mtile * 16 + ((lane < 16) ? 0 : 8);
    const int n  = ntile * 16 + l15;
#pragma unroll
    for (int r = 0; r < 8; ++r) op[(m0 + r) * kO + n] = acc[r];
  }
}

extern "C" void kernel_launch(void* const* d_in, const int* in_sizes, int n_in,
                              void* d_out, int out_size, void* d_ws, size_t ws_size,
                              hipStream_t stream) {
  (void)in_sizes; (void)n_in; (void)out_size; (void)d_ws; (void)ws_size;
  const float* input    = (const float*)d_in[0];
  const int*   size_raw = (const int*)d_in[1];
  const int*   pix      = (const int*)d_in[2];
  const int*   channels = (const int*)d_in[3];
  float*       out      = (float*)d_out;

  dim3 grid(kP * kS * kNC);  // 4096 blocks: one per (p, s, channel)
  dim3 block(128);           // 4 waves (wave32)
  hipLaunchKernelGGL(patch_sampler2d_wmma, grid, block, 0, stream,
                     input, size_raw, pix, channels, out);
}